// FeatureHead_25426206392368
// MI455X (gfx1250) — compile-verified
//
#include <hip/hip_runtime.h>
#include <math.h>

#define HH 512
#define WW 512
#define HWSZ (HH*WW)
#define OUTR 900
#define KMAX 10
#define NCLS 100

typedef float v2f __attribute__((ext_vector_type(2)));
typedef float v8f __attribute__((ext_vector_type(8)));

__device__ __forceinline__ float lrelu(float x){ return x > 0.f ? x : 0.01f*x; }
__device__ __forceinline__ float getpix(const float* __restrict__ p, int y, int x){
  return (y>=0 && y<HH && x>=0 && x<WW) ? p[y*WW+x] : 0.f;
}

// ---------------------------------------------------------------- zero fill
__global__ void k_zero(float4* __restrict__ p, long n4){
  long i = (long)blockIdx.x*blockDim.x + threadIdx.x;
  if(i < n4) p[i] = make_float4(0.f,0.f,0.f,0.f);
}

// ------------------------------------------- t = conv3x3(color, score_w)+b
__global__ void k_score_t(const float* __restrict__ color,
                          const float* __restrict__ sw, const float* __restrict__ sb,
                          float* __restrict__ t){
  int i = blockIdx.x*blockDim.x + threadIdx.x;
  if(i >= HWSZ) return;
  int y = i >> 9, x = i & 511;
  float acc = sb[0];
  for(int c=0;c<3;c++)
    for(int ky=0;ky<3;ky++)
      for(int kx=0;kx<3;kx++)
        acc += getpix(color + c*HWSZ, y+ky-1, x+kx-1) * sw[c*9 + ky*3 + kx];
  t[i] = acc;
}

// ---------------------------------- score = sigmoid(laplace(t)); out ch0 too
__global__ void k_score(const float* __restrict__ t,
                        float* __restrict__ scoreBuf, float* __restrict__ out0){
  int i = blockIdx.x*blockDim.x + threadIdx.x;
  if(i >= HWSZ) return;
  int y = i >> 9, x = i & 511;
  float v = 4.f*t[i] - getpix(t,y-1,x) - getpix(t,y+1,x) - getpix(t,y,x-1) - getpix(t,y,x+1);
  float s = 1.f / (1.f + __expf(-v));
  scoreBuf[i] = s;
  out0[i] = s;
}

// ---------------- wh0 = bbx_c2(lrelu(bbx_c1(laplace_per_channel(color))))
__global__ void k_wh0(const float* __restrict__ color,
                      const float* __restrict__ c1w, const float* __restrict__ c1b,
                      const float* __restrict__ c2w, const float* __restrict__ c2b,
                      float* __restrict__ wh0){
  int i = blockIdx.x*blockDim.x + threadIdx.x;
  if(i >= HWSZ) return;
  int y = i >> 9, x = i & 511;
  float lap[3];
  for(int c=0;c<3;c++){
    const float* p = color + c*HWSZ;
    lap[c] = 4.f*p[i] - getpix(p,y-1,x) - getpix(p,y+1,x) - getpix(p,y,x-1) - getpix(p,y,x+1);
  }
  float h6[6];
  #pragma unroll
  for(int j=0;j<6;j++)
    h6[j] = lrelu(c1b[j] + c1w[j*3+0]*lap[0] + c1w[j*3+1]*lap[1] + c1w[j*3+2]*lap[2]);
  float o0 = c2b[0], o1 = c2b[1];
  #pragma unroll
  for(int j=0;j<6;j++){ o0 += c2w[j]*h6[j]; o1 += c2w[6+j]*h6[j]; }
  wh0[i] = o0; wh0[HWSZ+i] = o1;
}

// ---------------- wh = ft_c2(lrelu(ft_c1(concat[score, wh0], 3x3 pad1)))
__global__ void k_wh(const float* __restrict__ score, const float* __restrict__ wh0,
                     const float* __restrict__ f1w, const float* __restrict__ f1b,
                     const float* __restrict__ f2w, const float* __restrict__ f2b,
                     float* __restrict__ wh){
  int i = blockIdx.x*blockDim.x + threadIdx.x;
  if(i >= HWSZ) return;
  int y = i >> 9, x = i & 511;
  float h6[6];
  #pragma unroll
  for(int j=0;j<6;j++) h6[j] = f1b[j];
  for(int c=0;c<3;c++){
    const float* plane = (c==0) ? score : (wh0 + (c-1)*HWSZ);
    for(int ky=0;ky<3;ky++)
      for(int kx=0;kx<3;kx++){
        float v = getpix(plane, y+ky-1, x+kx-1);
        #pragma unroll
        for(int j=0;j<6;j++) h6[j] += v * f1w[j*27 + c*9 + ky*3 + kx];
      }
  }
  #pragma unroll
  for(int j=0;j<6;j++) h6[j] = lrelu(h6[j]);
  float o0 = f2b[0], o1 = f2b[1];
  #pragma unroll
  for(int j=0;j<6;j++){ o0 += f2w[j]*h6[j]; o1 += f2w[6+j]*h6[j]; }
  wh[i] = o0; wh[HWSZ+i] = o1;
}

// ------------ top-10 argmax, ties -> lowest index (matches lax.top_k)
__global__ void k_topk(const float* __restrict__ score, int* __restrict__ topidx){
  __shared__ float rv[1024];
  __shared__ int   ri[1024];
  __shared__ int   top[KMAX];
  int tid = threadIdx.x;
  for(int k=0;k<KMAX;k++){
    float bv = -1e30f; int bi = 0x7fffffff;
    for(int i=tid;i<HWSZ;i+=1024){
      bool skip = false;
      for(int j=0;j<k;j++) skip |= (top[j] == i);
      if(skip) continue;
      float v = score[i];
      if(v > bv){ bv = v; bi = i; }       // ascending i -> first max kept
    }
    rv[tid] = bv; ri[tid] = bi; __syncthreads();
    for(int s=512;s>0;s>>=1){
      if(tid < s){
        float ov = rv[tid+s]; int oi = ri[tid+s];
        if(ov > rv[tid] || (ov == rv[tid] && oi < ri[tid])){ rv[tid]=ov; ri[tid]=oi; }
      }
      __syncthreads();
    }
    if(tid==0){ top[k] = ri[0]; topidx[k] = ri[0]; }
    __syncthreads();
  }
}

// ------- sel -> slot_idx/valid/boxes (stable: true idxs first, then false)
__global__ void k_select(const float* __restrict__ score, const float* __restrict__ wh,
                         const int* __restrict__ topidx,
                         int* __restrict__ slotidx, int* __restrict__ validout,
                         int* __restrict__ boxes){
  __shared__ int top[KMAX];
  __shared__ int cT[1024], cF[1024], pT[1024], pF[1024];
  __shared__ int listT[KMAX], listF[KMAX];
  __shared__ int totT;
  int tid = threadIdx.x;
  if(tid < KMAX) top[tid] = topidx[tid];
  __syncthreads();
  int base = tid * 256, ct = 0;
  for(int q=0;q<256;q++){
    int i = base + q;
    bool inT = false;
    for(int j=0;j<KMAX;j++) inT |= (top[j] == i);
    bool sel = (inT || score[i] > 0.8f) && (wh[i] > 5.f) && (wh[HWSZ+i] > 5.f);
    if(sel) ct++;
  }
  cT[tid] = ct; cF[tid] = 256 - ct; __syncthreads();
  if(tid == 0){
    int rT=0, rF=0;
    for(int u=0;u<1024;u++){ pT[u]=rT; pF[u]=rF; rT+=cT[u]; rF+=cF[u]; }
    totT = rT;
  }
  __syncthreads();
  int rT = pT[tid], rF = pF[tid];
  for(int q=0;q<256;q++){
    int i = base + q;
    bool inT = false;
    for(int j=0;j<KMAX;j++) inT |= (top[j] == i);
    bool sel = (inT || score[i] > 0.8f) && (wh[i] > 5.f) && (wh[HWSZ+i] > 5.f);
    if(sel){ if(rT < KMAX) listT[rT] = i; rT++; }
    else   { if(rF < KMAX) listF[rF] = i; rF++; }
  }
  __syncthreads();
  if(tid < KMAX){
    int nT = totT < KMAX ? totT : KMAX;
    int idx = (tid < nT) ? listT[tid] : listF[tid - nT];
    slotidx[tid] = idx;
    validout[tid] = (tid < totT) ? 1 : 0;
    float xs = (float)(idx & 511), ys = (float)(idx >> 9);
    float wv = wh[idx];
    float x1 = fminf(fmaxf(floorf(xs - wv), 0.f), (float)(WW-1));
    float x2 = fminf(fmaxf(ceilf (xs + wv), x1 + 1.f), (float)WW);
    float y1 = fminf(fmaxf(floorf(ys - wv), 0.f), (float)(HH-1));
    float y2 = fminf(fmaxf(ceilf (ys + wv), y1 + 1.f), (float)HH);
    boxes[tid*4+0] = (int)x1; boxes[tid*4+1] = (int)y1;
    boxes[tid*4+2] = (int)x2; boxes[tid*4+3] = (int)y2;
  }
}

// ------------- mask branch bs1..bs4 fully fused (strides==kernels -> no recompute)
// out m4: (10, 8, 100, 100)
__global__ void k_mask14(const float* __restrict__ mask, const int* __restrict__ boxes,
                         const float* __restrict__ w1, const float* __restrict__ b1,
                         const float* __restrict__ w2, const float* __restrict__ b2,
                         const float* __restrict__ w3, const float* __restrict__ b3,
                         const float* __restrict__ w4, const float* __restrict__ b4,
                         float* __restrict__ m4){
  __shared__ float sm[3184];
  float* s1w = sm;        float* s1b = sm + 288;
  float* s2w = sm + 320;  float* s2b = sm + 576;
  float* s3w = sm + 584;  float* s3b = sm + 2888;
  float* s4w = sm + 2920; float* s4b = sm + 3176;
  for(int i=threadIdx.x;i<288;i+=256)  s1w[i]=w1[i];
  for(int i=threadIdx.x;i<32;i+=256)   s1b[i]=b1[i];
  for(int i=threadIdx.x;i<256;i+=256)  s2w[i]=w2[i];
  if(threadIdx.x<8)                    s2b[threadIdx.x]=b2[threadIdx.x];
  for(int i=threadIdx.x;i<2304;i+=256) s3w[i]=w3[i];
  for(int i=threadIdx.x;i<32;i+=256)   s3b[i]=b3[i];
  for(int i=threadIdx.x;i<256;i+=256)  s4w[i]=w4[i];
  if(threadIdx.x<8)                    s4b[threadIdx.x]=b4[threadIdx.x];
  __syncthreads();
  int t = blockIdx.x*blockDim.x + threadIdx.x;
  if(t >= 10*100*100) return;
  int n = t / 10000, rem = t % 10000, oy = rem / 100, ox = rem % 100;
  int x1 = boxes[n*4+0], y1 = boxes[n*4+1];
  int dw = boxes[n*4+2] - x1, dh = boxes[n*4+3] - y1;
  int xs9[9], ys9[9];
  #pragma unroll
  for(int u=0;u<9;u++){
    xs9[u] = x1 + ((9*ox + u) * dw) / OUTR;
    ys9[u] = y1 + ((9*oy + u) * dh) / OUTR;
  }
  __builtin_prefetch(mask + ys9[0]*WW + xs9[0], 0, 1);
  float acc[32];
  #pragma unroll
  for(int c=0;c<32;c++) acc[c] = s3b[c];
  for(int qy=0;qy<3;qy++)
    for(int qx=0;qx<3;qx++){
      float pix[9];
      #pragma unroll
      for(int ky=0;ky<3;ky++)
        #pragma unroll
        for(int kx=0;kx<3;kx++)
          pix[ky*3+kx] = mask[ ys9[qy*3+ky]*WW + xs9[qx*3+kx] ];
      float v8[8];
      #pragma unroll
      for(int k=0;k<8;k++) v8[k] = s2b[k];
      for(int ch=0;ch<32;ch++){
        float a = s1b[ch];
        #pragma unroll
        for(int p=0;p<9;p++) a += pix[p]*s1w[ch*9+p];
        a = lrelu(a);
        #pragma unroll
        for(int k=0;k<8;k++) v8[k] += s2w[k*32+ch]*a;
      }
      int tap = qy*3 + qx;
      for(int c=0;c<32;c++){
        float a = 0.f;
        #pragma unroll
        for(int k=0;k<8;k++) a += s3w[c*72 + k*9 + tap]*v8[k];
        acc[c] += a;
      }
    }
  #pragma unroll
  for(int c=0;c<32;c++) acc[c] = lrelu(acc[c]);
  for(int k=0;k<8;k++){
    float a = s4b[k];
    #pragma unroll
    for(int c=0;c<32;c++) a += s4w[k*32+c]*acc[c];
    m4[(n*8+k)*10000 + oy*100 + ox] = a;
  }
}

// ------------- color branch cs1..cs4 fully fused. out c4: (10, 4, 100, 100)
__global__ void k_color14(const float* __restrict__ color, const int* __restrict__ boxes,
                          const float* __restrict__ w1, const float* __restrict__ b1,
                          const float* __restrict__ w2, const float* __restrict__ b2,
                          const float* __restrict__ w3, const float* __restrict__ b3,
                          const float* __restrict__ w4, const float* __restrict__ b4,
                          float* __restrict__ c4o){
  __shared__ float sm[2688];
  float* s1w = sm;        float* s1b = sm + 18;
  float* s2w = sm + 24;   float* s2b = sm + 48;
  float* s3w = sm + 52;   float* s3b = sm + 2644;
  float* s4w = sm + 2652; float* s4b = sm + 2684;
  if(threadIdx.x<18) s1w[threadIdx.x]=w1[threadIdx.x];
  if(threadIdx.x<6)  s1b[threadIdx.x]=b1[threadIdx.x];
  if(threadIdx.x<24) s2w[threadIdx.x]=w2[threadIdx.x];
  if(threadIdx.x<4)  s2b[threadIdx.x]=b2[threadIdx.x];
  for(int i=threadIdx.x;i<2592;i+=256) s3w[i]=w3[i];
  if(threadIdx.x<8)  s3b[threadIdx.x]=b3[threadIdx.x];
  if(threadIdx.x<32) s4w[threadIdx.x]=w4[threadIdx.x];
  if(threadIdx.x<4)  s4b[threadIdx.x]=b4[threadIdx.x];
  __syncthreads();
  int t = blockIdx.x*blockDim.x + threadIdx.x;
  if(t >= 10*100*100) return;
  int n = t / 10000, rem = t % 10000, oy = rem / 100, ox = rem % 100;
  int x1 = boxes[n*4+0], y1 = boxes[n*4+1];
  int dw = boxes[n*4+2] - x1, dh = boxes[n*4+3] - y1;
  int xs9[9], ys9[9];
  #pragma unroll
  for(int u=0;u<9;u++){
    xs9[u] = x1 + ((9*ox + u) * dw) / OUTR;
    ys9[u] = y1 + ((9*oy + u) * dh) / OUTR;
  }
  __builtin_prefetch(color + ys9[0]*WW + xs9[0], 0, 1);
  float acc[8];
  #pragma unroll
  for(int c=0;c<8;c++) acc[c] = s3b[c];
  for(int ty=0;ty<9;ty++)
    for(int tx=0;tx<9;tx++){
      int off = ys9[ty]*WW + xs9[tx];
      float r = color[off], g = color[HWSZ+off], b = color[2*HWSZ+off];
      float h6[6];
      #pragma unroll
      for(int j=0;j<6;j++)
        h6[j] = lrelu(s1b[j] + s1w[j*3]*r + s1w[j*3+1]*g + s1w[j*3+2]*b);
      float v4[4];
      #pragma unroll
      for(int k=0;k<4;k++){
        float a = s2b[k];
        #pragma unroll
        for(int j=0;j<6;j++) a += s2w[k*6+j]*h6[j];
        v4[k] = a;
      }
      int tap = ty*9 + tx;
      #pragma unroll
      for(int c=0;c<8;c++){
        float a = 0.f;
        #pragma unroll
        for(int k=0;k<4;k++) a += s3w[c*324 + k*81 + tap]*v4[k];
        acc[c] += a;
      }
    }
  #pragma unroll
  for(int c=0;c<8;c++) acc[c] = lrelu(acc[c]);
  for(int k2=0;k2<4;k2++){
    float a = s4b[k2];
    #pragma unroll
    for(int c=0;c<8;c++) a += s4w[k2*8+c]*acc[c];
    c4o[(n*4+k2)*10000 + oy*100 + ox] = a;
  }
}

// ------------- bs5+bs6 fused: (10,8,100,100) -> (10,8,25,25)
__global__ void k_bs56(const float* __restrict__ m4,
                       const float* __restrict__ w5, const float* __restrict__ b5,
                       const float* __restrict__ w6, const float* __restrict__ b6,
                       float* __restrict__ m6){
  __shared__ float sm[4392];
  float* s5w = sm; float* s5b = sm + 4096; float* s6w = sm + 4128; float* s6b = sm + 4384;
  for(int i=threadIdx.x;i<4096;i+=256) s5w[i]=w5[i];
  if(threadIdx.x<32)  s5b[threadIdx.x]=b5[threadIdx.x];
  for(int i=threadIdx.x;i<256;i+=256)  s6w[i]=w6[i];
  if(threadIdx.x<8)   s6b[threadIdx.x]=b6[threadIdx.x];
  __syncthreads();
  int t = blockIdx.x*blockDim.x + threadIdx.x;
  if(t >= 10*25*25) return;
  int n = t / 625, rem = t % 625, oy = rem / 25, ox = rem % 25;
  float acc[32];
  #pragma unroll
  for(int c=0;c<32;c++) acc[c] = s5b[c];
  for(int ty=0;ty<4;ty++)
    for(int tx=0;tx<4;tx++){
      int iy = 4*oy + ty, ix = 4*ox + tx;
      float v8[8];
      #pragma unroll
      for(int k=0;k<8;k++) v8[k] = m4[(n*8+k)*10000 + iy*100 + ix];
      int tap = ty*4 + tx;
      for(int c=0;c<32;c++){
        float a = 0.f;
        #pragma unroll
        for(int k=0;k<8;k++) a += s5w[c*128 + k*16 + tap]*v8[k];
        acc[c] += a;
      }
    }
  #pragma unroll
  for(int c=0;c<32;c++) acc[c] = lrelu(acc[c]);
  for(int k=0;k<8;k++){
    float a = s6b[k];
    #pragma unroll
    for(int c=0;c<32;c++) a += s6w[k*32+c]*acc[c];
    m6[(n*8+k)*625 + oy*25 + ox] = a;
  }
}

// ------------- cs5+cs6 fused: (10,4,100,100) -> (10,4,25,25)
__global__ void k_cs56(const float* __restrict__ c4i,
                       const float* __restrict__ w5, const float* __restrict__ b5,
                       const float* __restrict__ w6, const float* __restrict__ b6,
                       float* __restrict__ c6){
  __shared__ float sm[556];
  float* s5w = sm; float* s5b = sm + 512; float* s6w = sm + 520; float* s6b = sm + 552;
  for(int i=threadIdx.x;i<512;i+=256) s5w[i]=w5[i];
  if(threadIdx.x<8)  s5b[threadIdx.x]=b5[threadIdx.x];
  if(threadIdx.x<32) s6w[threadIdx.x]=w6[threadIdx.x];
  if(threadIdx.x<4)  s6b[threadIdx.x]=b6[threadIdx.x];
  __syncthreads();
  int t = blockIdx.x*blockDim.x + threadIdx.x;
  if(t >= 10*25*25) return;
  int n = t / 625, rem = t % 625, oy = rem / 25, ox = rem % 25;
  float acc[8];
  #pragma unroll
  for(int c=0;c<8;c++) acc[c] = s5b[c];
  for(int ty=0;ty<4;ty++)
    for(int tx=0;tx<4;tx++){
      int iy = 4*oy + ty, ix = 4*ox + tx;
      float v4[4];
      #pragma unroll
      for(int k=0;k<4;k++) v4[k] = c4i[(n*4+k)*10000 + iy*100 + ix];
      int tap = ty*4 + tx;
      #pragma unroll
      for(int c=0;c<8;c++){
        float a = 0.f;
        #pragma unroll
        for(int k=0;k<4;k++) a += s5w[c*64 + k*16 + tap]*v4[k];
        acc[c] += a;
      }
    }
  #pragma unroll
  for(int c=0;c<8;c++) acc[c] = lrelu(acc[c]);
  for(int k2=0;k2<4;k2++){
    float a = s6b[k2];
    #pragma unroll
    for(int c=0;c<8;c++) a += s6w[k2*8+c]*acc[c];
    c6[(n*4+k2)*625 + oy*25 + ox] = a;
  }
}

// ------------- bs7+bs8 fused: (10,8,25,25) -> (10,16,5,5)
__global__ void k_bs78(const float* __restrict__ m6,
                       const float* __restrict__ w7, const float* __restrict__ b7,
                       const float* __restrict__ w8, const float* __restrict__ b8,
                       float* __restrict__ m8){
  __shared__ float sm[6960];
  float* s7w = sm; float* s7b = sm + 6400; float* s8w = sm + 6432; float* s8b = sm + 6944;
  for(int i=threadIdx.x;i<6400;i+=256) s7w[i]=w7[i];
  if(threadIdx.x<32)  s7b[threadIdx.x]=b7[threadIdx.x];
  for(int i=threadIdx.x;i<512;i+=256)  s8w[i]=w8[i];
  if(threadIdx.x<16)  s8b[threadIdx.x]=b8[threadIdx.x];
  __syncthreads();
  int t = blockIdx.x*blockDim.x + threadIdx.x;
  if(t >= 10*5*5) return;
  int n = t / 25, rem = t % 25, oy = rem / 5, ox = rem % 5;
  float acc[32];
  #pragma unroll
  for(int c=0;c<32;c++) acc[c] = s7b[c];
  for(int ty=0;ty<5;ty++)
    for(int tx=0;tx<5;tx++){
      int iy = 5*oy + ty, ix = 5*ox + tx;
      float v8[8];
      #pragma unroll
      for(int k=0;k<8;k++) v8[k] = m6[(n*8+k)*625 + iy*25 + ix];
      int tap = ty*5 + tx;
      for(int c=0;c<32;c++){
        float a = 0.f;
        #pragma unroll
        for(int k=0;k<8;k++) a += s7w[c*200 + k*25 + tap]*v8[k];
        acc[c] += a;
      }
    }
  #pragma unroll
  for(int c=0;c<32;c++) acc[c] = lrelu(acc[c]);
  for(int k=0;k<16;k++){
    float a = s8b[k];
    #pragma unroll
    for(int c=0;c<32;c++) a += s8w[k*32+c]*acc[c];
    m8[(n*16+k)*25 + oy*5 + ox] = a;
  }
}

// ------------- cs7+cs8 fused: (10,4,25,25) -> (10,4,5,5)
__global__ void k_cs78(const float* __restrict__ c6,
                       const float* __restrict__ w7, const float* __restrict__ b7,
                       const float* __restrict__ w8, const float* __restrict__ b8,
                       float* __restrict__ c8){
  __shared__ float sm[844];
  float* s7w = sm; float* s7b = sm + 800; float* s8w = sm + 808; float* s8b = sm + 840;
  for(int i=threadIdx.x;i<800;i+=256) s7w[i]=w7[i];
  if(threadIdx.x<8)  s7b[threadIdx.x]=b7[threadIdx.x];
  if(threadIdx.x<32) s8w[threadIdx.x]=w8[threadIdx.x];
  if(threadIdx.x<4)  s8b[threadIdx.x]=b8[threadIdx.x];
  __syncthreads();
  int t = blockIdx.x*blockDim.x + threadIdx.x;
  if(t >= 10*5*5) return;
  int n = t / 25, rem = t % 25, oy = rem / 5, ox = rem % 5;
  float acc[8];
  #pragma unroll
  for(int c=0;c<8;c++) acc[c] = s7b[c];
  for(int ty=0;ty<5;ty++)
    for(int tx=0;tx<5;tx++){
      int iy = 5*oy + ty, ix = 5*ox + tx;
      float v4[4];
      #pragma unroll
      for(int k=0;k<4;k++) v4[k] = c6[(n*4+k)*625 + iy*25 + ix];
      int tap = ty*5 + tx;
      #pragma unroll
      for(int c=0;c<8;c++){
        float a = 0.f;
        #pragma unroll
        for(int k=0;k<4;k++) a += s7w[c*100 + k*25 + tap]*v4[k];
        acc[c] += a;
      }
    }
  #pragma unroll
  for(int c=0;c<8;c++) acc[c] = lrelu(acc[c]);
  for(int k2=0;k2<4;k2++){
    float a = s8b[k2];
    #pragma unroll
    for(int c=0;c<8;c++) a += s8w[k2*8+c]*acc[c];
    c8[(n*4+k2)*25 + oy*5 + ox] = a;
  }
}

// -------- cl1 + pack X[K=752][N=16]: X[o*25+pos][crop] = lrelu(cl1(comb))
__global__ void k_head_stage(const float* __restrict__ m8, const float* __restrict__ c8,
                             const float* __restrict__ w1, const float* __restrict__ b1,
                             float* __restrict__ X){
  int t = blockIdx.x*blockDim.x + threadIdx.x;
  if(t >= 752*16) return;
  int k = t / 16, n = t % 16;
  float val = 0.f;
  if(n < 10 && k < 750){
    int o = k / 25, pos = k % 25;
    float a = b1[o];
    #pragma unroll
    for(int i=0;i<20;i++){
      float f = (i < 16) ? m8[(n*16+i)*25 + pos] : c8[(n*4+(i-16))*25 + pos];
      a += w1[o*20+i]*f;
    }
    val = lrelu(a);
  }
  X[k*16+n] = val;
}

// -------- cl2 as WMMA GEMM: pred[10x100] = W2[100x750] * X[750x16(pad)] + b2
// A 16x4 f32 layout (ISA 7.12.2): lane<16 -> M=lane, K pair {k,k+1};
// lane>=16 -> M=lane-16, K pair {k+2,k+3}. B 4x16 mirrored. D: 8 VGPRs,
// lanes 0-15 rows 0-7, lanes 16-31 rows 8-15.
__global__ void k_wmma_head(const float* __restrict__ W2, const float* __restrict__ X,
                            const float* __restrict__ b2, float* __restrict__ pred){
  int lane  = threadIdx.x;
  int mtile = blockIdx.x;                 // 0..6 (7*16 = 112 >= 100)
  int mrow  = mtile*16 + (lane & 15);
  int khalf = (lane >> 4) * 2;
  int ncol  = lane & 15;
  v8f acc = {};
  for(int k0 = 0; k0 < 752; k0 += 8){
    #pragma unroll
    for(int s = 0; s < 2; s++){
      int kb  = k0 + s*4 + khalf;
      int mi  = (mrow < 100) ? mrow : 0;
      int ka0 = (kb   < 750) ? kb   : 0;
      int ka1 = (kb+1 < 750) ? kb+1 : 0;
      float g0 = (mrow < 100 && kb   < 750) ? 1.f : 0.f;
      float g1 = (mrow < 100 && kb+1 < 750) ? 1.f : 0.f;
      v2f a, b;
      a[0] = W2[mi*750 + ka0] * g0;
      a[1] = W2[mi*750 + ka1] * g1;
      b[0] = X[(kb  )*16 + ncol];
      b[1] = X[(kb+1)*16 + ncol];
      acc = __builtin_amdgcn_wmma_f32_16x16x4_f32(
          /*neg_a=*/false, a, /*neg_b=*/false, b,
          /*c_mod=*/(short)0, acc, /*reuse_a=*/false, /*reuse_b=*/false);
    }
  }
  int rbase = mtile*16 + ((lane >> 4) ? 8 : 0);
  #pragma unroll
  for(int j=0;j<8;j++){
    int r = rbase + j;
    if(r < 100 && ncol < 10) pred[ncol*100 + r] = acc[j] + b2[r];
  }
}

// -------- scatter predictions into cls_grid channels (out ch 1..100)
__global__ void k_scatter(const float* __restrict__ pred, const int* __restrict__ slotidx,
                          const int* __restrict__ valid, float* __restrict__ out){
  int t = blockIdx.x*blockDim.x + threadIdx.x;
  if(t >= KMAX*NCLS) return;
  int s = t / NCLS, c = t % NCLS;
  if(valid[s]) out[(1+c)*HWSZ + slotidx[s]] = pred[s*NCLS + c];
}

// ---------------------------------------------------------------- launcher
extern "C" void kernel_launch(void* const* d_in, const int* in_sizes, int n_in,
                              void* d_out, int out_size, void* d_ws, size_t ws_size,
                              hipStream_t stream){
  const float* mask     = (const float*)d_in[0];
  const float* color    = (const float*)d_in[1];
  const float* bbx_c1_w = (const float*)d_in[2];
  const float* bbx_c1_b = (const float*)d_in[3];
  const float* bbx_c2_w = (const float*)d_in[4];
  const float* bbx_c2_b = (const float*)d_in[5];
  const float* score_w  = (const float*)d_in[6];
  const float* score_b  = (const float*)d_in[7];
  const float* ft_c1_w  = (const float*)d_in[8];
  const float* ft_c1_b  = (const float*)d_in[9];
  const float* ft_c2_w  = (const float*)d_in[10];
  const float* ft_c2_b  = (const float*)d_in[11];
  const float* bs1_w = (const float*)d_in[12]; const float* bs1_b = (const float*)d_in[13];
  const float* bs2_w = (const float*)d_in[14]; const float* bs2_b = (const float*)d_in[15];
  const float* bs3_w = (const float*)d_in[16]; const float* bs3_b = (const float*)d_in[17];
  const float* bs4_w = (const float*)d_in[18]; const float* bs4_b = (const float*)d_in[19];
  const float* bs5_w = (const float*)d_in[20]; const float* bs5_b = (const float*)d_in[21];
  const float* bs6_w = (const float*)d_in[22]; const float* bs6_b = (const float*)d_in[23];
  const float* bs7_w = (const float*)d_in[24]; const float* bs7_b = (const float*)d_in[25];
  const float* bs8_w = (const float*)d_in[26]; const float* bs8_b = (const float*)d_in[27];
  const float* cs1_w = (const float*)d_in[28]; const float* cs1_b = (const float*)d_in[29];
  const float* cs2_w = (const float*)d_in[30]; const float* cs2_b = (const float*)d_in[31];
  const float* cs3_w = (const float*)d_in[32]; const float* cs3_b = (const float*)d_in[33];
  const float* cs4_w = (const float*)d_in[34]; const float* cs4_b = (const float*)d_in[35];
  const float* cs5_w = (const float*)d_in[36]; const float* cs5_b = (const float*)d_in[37];
  const float* cs6_w = (const float*)d_in[38]; const float* cs6_b = (const float*)d_in[39];
  const float* cs7_w = (const float*)d_in[40]; const float* cs7_b = (const float*)d_in[41];
  const float* cs8_w = (const float*)d_in[42]; const float* cs8_b = (const float*)d_in[43];
  const float* cl1_w = (const float*)d_in[44]; const float* cl1_b = (const float*)d_in[45];
  const float* cl2_w = (const float*)d_in[46]; const float* cl2_b = (const float*)d_in[47];

  float* out = (float*)d_out;
  float* wsf = (float*)d_ws;

  // workspace layout (floats)
  float* tBuf     = wsf + 0;        // 262144
  float* scoreBuf = wsf + 262144;   // 262144
  float* wh0Buf   = wsf + 524288;   // 524288
  float* whBuf    = wsf + 1048576;  // 524288
  float* m4Buf    = wsf + 1572864;  // 800000
  float* c4Buf    = wsf + 2372864;  // 400000
  float* m6Buf    = wsf + 2772864;  // 50000
  float* c6Buf    = wsf + 2822864;  // 25000
  float* m8Buf    = wsf + 2847864;  // 4000
  float* c8Buf    = wsf + 2851864;  // 1000
  float* XBuf     = wsf + 2852864;  // 12032
  float* predBuf  = wsf + 2864896;  // 1000
  int*   iBuf     = (int*)(wsf + 2865896);
  int* topidx  = iBuf;       // 10
  int* slotidx = iBuf + 10;  // 10
  int* validB  = iBuf + 20;  // 10
  int* boxes   = iBuf + 30;  // 40

  // 1) zero the 100 cls_grid channels (105 MB -> dominates; float4 stores)
  long n4 = (long)NCLS * HWSZ / 4;
  k_zero<<<(int)((n4 + 255) / 256), 256, 0, stream>>>((float4*)(out + HWSZ), n4);

  // 2) score / bbx path over 512x512
  k_score_t<<<HWSZ/256, 256, 0, stream>>>(color, score_w, score_b, tBuf);
  k_score  <<<HWSZ/256, 256, 0, stream>>>(tBuf, scoreBuf, out);
  k_wh0    <<<HWSZ/256, 256, 0, stream>>>(color, bbx_c1_w, bbx_c1_b, bbx_c2_w, bbx_c2_b, wh0Buf);
  k_wh     <<<HWSZ/256, 256, 0, stream>>>(scoreBuf, wh0Buf, ft_c1_w, ft_c1_b, ft_c2_w, ft_c2_b, whBuf);

  // 3) deterministic top-k + slot selection + boxes
  k_topk  <<<1, 1024, 0, stream>>>(scoreBuf, topidx);
  k_select<<<1, 1024, 0, stream>>>(scoreBuf, whBuf, topidx, slotidx, validB, boxes);

  // 4) classifier (crop gather fused into first layer; all strided convs
  //    have kernel==stride so full fusion has zero recompute)
  k_mask14 <<<(100000+255)/256, 256, 0, stream>>>(mask, boxes,
              bs1_w, bs1_b, bs2_w, bs2_b, bs3_w, bs3_b, bs4_w, bs4_b, m4Buf);
  k_color14<<<(100000+255)/256, 256, 0, stream>>>(color, boxes,
              cs1_w, cs1_b, cs2_w, cs2_b, cs3_w, cs3_b, cs4_w, cs4_b, c4Buf);
  k_bs56<<<(6250+255)/256, 256, 0, stream>>>(m4Buf, bs5_w, bs5_b, bs6_w, bs6_b, m6Buf);
  k_cs56<<<(6250+255)/256, 256, 0, stream>>>(c4Buf, cs5_w, cs5_b, cs6_w, cs6_b, c6Buf);
  k_bs78<<<1, 256, 0, stream>>>(m6Buf, bs7_w, bs7_b, bs8_w, bs8_b, m8Buf);
  k_cs78<<<1, 256, 0, stream>>>(c6Buf, cs7_w, cs7_b, cs8_w, cs8_b, c8Buf);

  // 5) head: cl1 pack + cl2 as WMMA f32 GEMM, then scatter
  k_head_stage<<<(752*16 + 255)/256, 256, 0, stream>>>(m8Buf, c8Buf, cl1_w, cl1_b, XBuf);
  k_wmma_head<<<7, 32, 0, stream>>>(cl2_w, XBuf, cl2_b, predBuf);
  k_scatter<<<(KMAX*NCLS + 255)/256, 256, 0, stream>>>(predBuf, slotidx, validB, out);
}